// GCN_22668837388733
// MI455X (gfx1250) — compile-verified
//
#include <hip/hip_runtime.h>
#include <hip/hip_bf16.h>

typedef __attribute__((ext_vector_type(2))) float v2f;
typedef __attribute__((ext_vector_type(8))) float v8f;

#define IN_CH  128
#define HID_CH 128
#define OUT_CH 64

// ---------------------------------------------------------------------------
// Utility kernels
// ---------------------------------------------------------------------------
__global__ void zero_kernel(float* __restrict__ p, long long n) {
    long long i = (long long)blockIdx.x * blockDim.x + threadIdx.x;
    long long stride = (long long)gridDim.x * blockDim.x;
    for (; i < n; i += stride) p[i] = 0.0f;
}

__global__ void cast_edges_kernel(const long long* __restrict__ ei,
                                  int* __restrict__ src, int* __restrict__ dst,
                                  long long E) {
    long long i = (long long)blockIdx.x * blockDim.x + threadIdx.x;
    if (i < E) {
        src[i] = (int)ei[i];
        dst[i] = (int)ei[E + i];
    }
}

__global__ void deg_kernel(const int* __restrict__ dst, float* __restrict__ deg,
                           long long E) {
    long long i = (long long)blockIdx.x * blockDim.x + threadIdx.x;
    if (i < E) unsafeAtomicAdd(&deg[dst[i]], 1.0f);
}

__global__ void dinv_kernel(const float* __restrict__ deg, float* __restrict__ dinv,
                            int N) {
    int i = blockIdx.x * blockDim.x + threadIdx.x;
    if (i < N) dinv[i] = rsqrtf(deg[i] + 1.0f);  // +1 for self-loop; always > 0
}

// ---------------------------------------------------------------------------
// fp32 WMMA GEMM:  out[nrows x NCOL] = A[nrows x K] * W[K x NCOL]
// 256 threads = 8 waves. Each wave computes one 16x16 tile via
// V_WMMA_F32_16X16X4_F32, K-loop in steps of 4. W is staged in LDS.
//
// Fragment layouts per CDNA5 ISA 7.12.2:
//   A (16x4, 2 VGPRs):  lanes 0-15 -> M=lane, K={0,1}; lanes 16-31 -> K={2,3}
//   B (4x16, 2 VGPRs):  lanes 0-15 -> N=lane, rows {0,1}; lanes 16-31 -> {2,3}
//   C (16x16, 8 VGPRs): VGPR i: lanes 0-15 -> M=i, lanes 16-31 -> M=8+i, N=lane&15
// ---------------------------------------------------------------------------
template <int K, int NCOL>
__global__ __launch_bounds__(256) void gemm_wmma(const float* __restrict__ A,
                                                 const float* __restrict__ W,
                                                 float* __restrict__ out,
                                                 int nrows) {
    constexpr int WAVES    = 8;
    constexpr int WPRT     = NCOL / 16;       // waves per row-tile (cols/16)
    constexpr int ROWTILES = WAVES / WPRT;    // 16-row tiles per block

    __shared__ float Ws[K * NCOL];
    for (int i = threadIdx.x; i < K * NCOL; i += 256) Ws[i] = W[i];
    __syncthreads();

    const int wave    = threadIdx.x >> 5;
    const int lane    = threadIdx.x & 31;
    const int halfsel = lane >> 4;            // 0: lanes 0-15, 1: lanes 16-31
    const int l15     = lane & 15;

    const int rowTile = wave / WPRT;
    const int colBase = (wave % WPRT) * 16;
    const int rowBase = (blockIdx.x * ROWTILES + rowTile) * 16;
    (void)nrows;  // grid sized so rowBase+15 < nrows (nrows % (16*ROWTILES) == 0)

    const float* Arow = A + (long long)(rowBase + l15) * K;

    v8f c = {};
#pragma unroll 4
    for (int k = 0; k < K; k += 4) {
        const int koff = k + halfsel * 2;
        v2f a = *(const v2f*)(Arow + koff);                 // K, K+1 (8B aligned)
        v2f b;
        b.x = Ws[(koff + 0) * NCOL + colBase + l15];
        b.y = Ws[(koff + 1) * NCOL + colBase + l15];
        c = __builtin_amdgcn_wmma_f32_16x16x4_f32(
                /*neg_a=*/false, a, /*neg_b=*/false, b,
                /*c_mod=*/(short)0, c, /*reuse_a=*/false, /*reuse_b=*/false);
    }

#pragma unroll
    for (int i = 0; i < 8; ++i) {
        const int row = rowBase + i + 8 * halfsel;
        out[(long long)row * NCOL + colBase + l15] = c[i];
    }
}

// ---------------------------------------------------------------------------
// Edge aggregation: one wave per edge; lane owns C/32 channels.
// Coalesced gather of h[src] (L2-resident) + fp32 global atomics to agg[dst].
// ---------------------------------------------------------------------------
template <int C>
__global__ __launch_bounds__(256) void agg_kernel(const float* __restrict__ h,
                                                  const int* __restrict__ src,
                                                  const int* __restrict__ dst,
                                                  const float* __restrict__ dinv,
                                                  float* __restrict__ agg,
                                                  long long E) {
    constexpr int VEC = C / 32;
    const long long tid = (long long)blockIdx.x * blockDim.x + threadIdx.x;
    const long long e   = tid >> 5;
    if (e >= E) return;
    const int lane = threadIdx.x & 31;

    const int s = src[e];
    const int d = dst[e];
    const float nrm = dinv[s] * dinv[d];

    const float* hp = h   + (long long)s * C + lane * VEC;
    float*       op = agg + (long long)d * C + lane * VEC;
#pragma unroll
    for (int j = 0; j < VEC; ++j) unsafeAtomicAdd(op + j, nrm * hp[j]);
}

// Epilogue 1: h2 = relu(agg1 + dinv^2 * h + b1), written in place over agg1.
__global__ void epi1_kernel(float* __restrict__ agg, const float* __restrict__ h,
                            const float* __restrict__ dinv,
                            const float* __restrict__ b1, int N) {
    long long i = (long long)blockIdx.x * blockDim.x + threadIdx.x;
    if (i >= (long long)N * HID_CH) return;
    const int node = (int)(i >> 7);
    const int ch   = (int)(i & (HID_CH - 1));
    const float di = dinv[node];
    const float v  = agg[i] + di * di * h[i] + b1[ch];
    agg[i] = v > 0.0f ? v : 0.0f;
}

// Epilogue 2: out = agg2 + dinv^2 * p + b2.
__global__ void epi2_kernel(const float* __restrict__ agg,
                            const float* __restrict__ p,
                            const float* __restrict__ dinv,
                            const float* __restrict__ b2,
                            float* __restrict__ out, int N) {
    long long i = (long long)blockIdx.x * blockDim.x + threadIdx.x;
    if (i >= (long long)N * OUT_CH) return;
    const int node = (int)(i >> 6);
    const int ch   = (int)(i & (OUT_CH - 1));
    const float di = dinv[node];
    out[i] = agg[i] + di * di * p[i] + b2[ch];
}

// ---------------------------------------------------------------------------
// Launcher
// ---------------------------------------------------------------------------
extern "C" void kernel_launch(void* const* d_in, const int* in_sizes, int n_in,
                              void* d_out, int out_size, void* d_ws, size_t ws_size,
                              hipStream_t stream) {
    const float*     x  = (const float*)d_in[0];
    const long long* ei = (const long long*)d_in[1];
    const float*     W1 = (const float*)d_in[2];
    const float*     b1 = (const float*)d_in[3];
    const float*     W2 = (const float*)d_in[4];
    const float*     b2 = (const float*)d_in[5];

    const int       N = in_sizes[0] / IN_CH;       // 100000 (divisible by 32)
    const long long E = (long long)in_sizes[1] / 2; // 3200000

    // Workspace layout (floats / ints, all 4B types; h offset is 16B aligned)
    float* deg  = (float*)d_ws;                        // N
    float* dinv = deg + N;                             // N
    int*   src  = (int*)(dinv + N);                    // E
    int*   dst  = src + E;                             // E
    float* h    = (float*)(dst + E);                   // N*128
    float* agg1 = h + (long long)N * HID_CH;           // N*128 (becomes h2)
    float* p    = agg1 + (long long)N * HID_CH;        // N*64
    float* agg2 = p + (long long)N * OUT_CH;           // N*64

    const int T = 256;

    // 1) zero deg + both accumulators
    zero_kernel<<<2048, T, 0, stream>>>(deg, N);
    zero_kernel<<<4096, T, 0, stream>>>(agg1, (long long)N * HID_CH);
    zero_kernel<<<4096, T, 0, stream>>>(agg2, (long long)N * OUT_CH);

    // 2) int64 -> int32 edge list
    cast_edges_kernel<<<(unsigned)((E + T - 1) / T), T, 0, stream>>>(ei, src, dst, E);

    // 3) degrees + inverse-sqrt norm
    deg_kernel<<<(unsigned)((E + T - 1) / T), T, 0, stream>>>(dst, deg, E);
    dinv_kernel<<<(N + T - 1) / T, T, 0, stream>>>(deg, dinv, N);

    // 4) layer 1: h = x @ W1   (WMMA f32, 16 rows x 128 cols per block)
    gemm_wmma<IN_CH, HID_CH><<<N / 16, T, 0, stream>>>(x, W1, h, N);

    // 5) layer-1 aggregation over edges (one wave per edge)
    const long long agg_threads = E * 32;
    agg_kernel<HID_CH><<<(unsigned)((agg_threads + T - 1) / T), T, 0, stream>>>(
        h, src, dst, dinv, agg1, E);

    // 6) epilogue: self-loop + bias + ReLU -> h2 (in place in agg1)
    epi1_kernel<<<(unsigned)(((long long)N * HID_CH + T - 1) / T), T, 0, stream>>>(
        agg1, h, dinv, b1, N);

    // 7) layer 2: p = h2 @ W2  (WMMA f32, 32 rows x 64 cols per block)
    gemm_wmma<HID_CH, OUT_CH><<<N / 32, T, 0, stream>>>(agg1, W2, p, N);

    // 8) layer-2 aggregation
    agg_kernel<OUT_CH><<<(unsigned)((agg_threads + T - 1) / T), T, 0, stream>>>(
        p, src, dst, dinv, agg2, E);

    // 9) final epilogue -> d_out
    epi2_kernel<<<(unsigned)(((long long)N * OUT_CH + T - 1) / T), T, 0, stream>>>(
        agg2, p, dinv, b2, (float*)d_out, N);
}